// Transform_55551107006941
// MI455X (gfx1250) — compile-verified
//
#include <hip/hip_runtime.h>
#include <cstdint>

typedef unsigned int  u32;
typedef unsigned short u16;

#define N_IMG 128
#define IMG_E 262144u            // 512*512
#define NBINS 8192
#define BPI   4                  // blocks per image in scan kernels
#define NBLK  (N_IMG*BPI)        // 512
#define EPB   65536              // elements per block (scan kernels)
#define K_SEL 3355443u           // int(0.1 * 128*512*512)
#define OUT_E 8388608u           // 128*256*256

// ---------------- workspace layout (bytes) ----------------
static constexpr size_t WS_BH      = 0;                                   // 512*8192*4  = 16 MiB (becomes cursor bases)
static constexpr size_t WS_IMGBASE = WS_BH      + (size_t)NBLK*NBINS*4;   // 128*8193*4
static constexpr size_t WS_KEYS    = WS_IMGBASE + (size_t)N_IMG*8193*4;   // 128*262144*2 = 64 MiB
static constexpr size_t WS_SELL2   = WS_KEYS    + (size_t)N_IMG*IMG_E*2;  // 65536*4
static constexpr size_t WS_TOTAL   = WS_SELL2   + (size_t)65536*4;        // 8192*4
static constexpr size_t WS_STATE   = WS_TOTAL   + (size_t)NBINS*4;        // [0]=sel1 [1]=krem [2]=sel2 [3]=Tbits

// ---------------- CDNA5 async global->LDS path ----------------
#if defined(__gfx1250__) && __has_builtin(__builtin_amdgcn_global_load_async_to_lds_b128) && __has_builtin(__builtin_amdgcn_s_wait_asynccnt)
#define USE_ASYNC 1
typedef int v4i __attribute__((ext_vector_type(4)));
typedef __attribute__((address_space(1))) v4i* gptr4;
typedef __attribute__((address_space(3))) v4i* lptr4;
__device__ __forceinline__ void async_ld16(const float* g, float* l) {
  __builtin_amdgcn_global_load_async_to_lds_b128((gptr4)(float*)g, (lptr4)l, 0, 0);
}
template<int N> __device__ __forceinline__ void wait_async_le() {
  __builtin_amdgcn_s_wait_asynccnt(N);
}
#else
#define USE_ASYNC 0
#if defined(__gfx1250__)
#warning "gfx1250 async-to-LDS builtins unavailable: falling back to plain loads"
#endif
#endif

// ---------------- CDNA5 WMMA path for the column-sum reduction ----------------
#if defined(__gfx1250__) && __has_builtin(__builtin_amdgcn_wmma_f32_16x16x4_f32)
#define USE_WMMA 1
typedef float v2f __attribute__((ext_vector_type(2)));
typedef float v8f __attribute__((ext_vector_type(8)));
#else
#define USE_WMMA 0
#if defined(__gfx1250__)
#warning "wmma_f32_16x16x4_f32 builtin unavailable: scalar sumhist fallback"
#endif
#endif

// bucket (13 bits) + in-bucket key (16 bits); all steps exact in fp32:
// v*8192 exact (pow2), g - c exact (suffix bits), frac*65536 exact (pow2).
__device__ __forceinline__ void bucket_key(float v, u32& c, u32& kv) {
  float g = v * 8192.0f;
  u32 ci = (u32)g; if (ci > 8191u) ci = 8191u;
  float frac = g - (float)ci;
  u32 k = (u32)(frac * 65536.0f); if (k > 65535u) k = 65535u;
  c = ci; kv = k;
}

// ---------------- kernels ----------------
__global__ void k_zero(u32* __restrict__ selL2, u32* __restrict__ state) {
  u32 i = blockIdx.x * 256u + threadIdx.x;
  if (i < 65536u) selL2[i] = 0u;
  if (i == 0u) state[3] = 0xFFFFFFFFu;   // max bit pattern for atomicMin on positive floats
}

// Pass 1: per-block 8192-bin histogram -> bh[block][bin] (no global atomics)
__global__ __launch_bounds__(256) void k_hist(const float* __restrict__ in, u32* __restrict__ bh) {
  __shared__ u32 hist[NBINS];
#if USE_ASYNC
  __shared__ __align__(16) float stage[8][2][128];
#endif
  for (int i = threadIdx.x; i < NBINS; i += 256) hist[i] = 0u;
  __syncthreads();
  const u32 lane = threadIdx.x & 31u, w = threadIdx.x >> 5;
  const float* src = in + (size_t)blockIdx.x * EPB + (size_t)w * 8192u;
#if USE_ASYNC
  {
    float* b0 = &stage[w][0][0];
    float* b1 = &stage[w][1][0];
    async_ld16(src + lane * 4u, b0 + lane * 4u);
    for (int t = 0; t < 64; ++t) {
      float* cur = (t & 1) ? b1 : b0;
      float* nxt = (t & 1) ? b0 : b1;
      if (t < 63) { async_ld16(src + (size_t)(t + 1) * 128u + lane * 4u, nxt + lane * 4u); wait_async_le<1>(); }
      else        { wait_async_le<0>(); }
      const float4 v = *(const float4*)(cur + lane * 4u);
      u32 c, k;
      bucket_key(v.x, c, k); atomicAdd(&hist[c], 1u);
      bucket_key(v.y, c, k); atomicAdd(&hist[c], 1u);
      bucket_key(v.z, c, k); atomicAdd(&hist[c], 1u);
      bucket_key(v.w, c, k); atomicAdd(&hist[c], 1u);
    }
  }
#else
  for (int t = 0; t < 64; ++t) {
    const float4 v = *(const float4*)(src + (size_t)t * 128u + lane * 4u);
    u32 c, k;
    bucket_key(v.x, c, k); atomicAdd(&hist[c], 1u);
    bucket_key(v.y, c, k); atomicAdd(&hist[c], 1u);
    bucket_key(v.z, c, k); atomicAdd(&hist[c], 1u);
    bucket_key(v.w, c, k); atomicAdd(&hist[c], 1u);
  }
#endif
  __syncthreads();
  u32* dst = bh + (size_t)blockIdx.x * NBINS;
  for (int i = threadIdx.x; i < NBINS; i += 256) dst[i] = hist[i];
}

// Per-image exclusive prefix (imgbase[8193]) and in-place rewrite of bh -> per-block cursor bases
__global__ __launch_bounds__(256) void k_prefix(u32* __restrict__ bh, u32* __restrict__ imgbase) {
  __shared__ u32 bc[NBINS];
  __shared__ u32 psum[257];
  const int img = blockIdx.x;
  const int t = threadIdx.x;
  u32* r0 = bh + ((size_t)img * BPI + 0) * NBINS;
  u32* r1 = bh + ((size_t)img * BPI + 1) * NBINS;
  u32* r2 = bh + ((size_t)img * BPI + 2) * NBINS;
  u32* r3 = bh + ((size_t)img * BPI + 3) * NBINS;
  u32 s = 0;
  for (int j = 0; j < 32; ++j) {
    int c = t * 32 + j;
    u32 v = r0[c] + r1[c] + r2[c] + r3[c];
    bc[c] = v; s += v;
  }
  psum[t] = s;
  __syncthreads();
  if (t == 0) {
    u32 run = 0;
    for (int i = 0; i < 256; ++i) { u32 x = psum[i]; psum[i] = run; run += x; }
    psum[256] = run;
  }
  __syncthreads();
  u32 run = psum[t];
  u32* ib = imgbase + (size_t)img * 8193u;
  for (int j = 0; j < 32; ++j) {
    int c = t * 32 + j;
    u32 b0 = r0[c], b1 = r1[c], b2 = r2[c];
    ib[c] = run;
    r0[c] = run;
    r1[c] = run + b0;
    r2[c] = run + b0 + b1;
    r3[c] = run + b0 + b1 + b2;
    run += bc[c];
  }
  if (t == 255) ib[8192] = psum[256];
}

// Global histogram = sum over 128 images of per-image bin counts.
// WMMA formulation: total(1..16 x 16bins) = ones(16x4) * counts(4imgs x 16bins), accumulated
// over 32 chunks of 4 images. All-ones A makes the k-slot permutation in B's layout irrelevant.
// Counts are exact in f32 (<< 2^24). Grid: 64 blocks x 256 threads; wave handles 16 bins.
__global__ __launch_bounds__(256) void k_sumhist(const u32* __restrict__ imgbase, u32* __restrict__ total) {
#if USE_WMMA
  const u32 lane = threadIdx.x & 31u;
  const u32 wave = threadIdx.x >> 5;
  const u32 bin0 = blockIdx.x * 128u + wave * 16u;
  const u32 n = lane & 15u;
  const u32 khalf = lane >> 4;          // lanes 16-31 supply the other two K rows
  v2f a; a.x = 1.0f; a.y = 1.0f;        // A = ones(16x4)
  v8f c = {};
  for (int img = 0; img < N_IMG; img += 4) {
    const u32* p0 = imgbase + (size_t)(img + khalf * 2u) * 8193u + bin0 + n;
    const u32* p1 = p0 + 8193u;
    v2f b;
    b.x = (float)(p0[1] - p0[0]);
    b.y = (float)(p1[1] - p1[0]);
    c = __builtin_amdgcn_wmma_f32_16x16x4_f32(false, a, false, b, (short)0, c, false, false);
  }
  if (lane < 16u) total[bin0 + n] = (u32)c[0];   // D row M=0: lanes 0-15, VGPR0
#else
  int cbin = blockIdx.x * 128 + (threadIdx.x & 127);  // duplicate writes of identical values are benign
  u32 s = 0;
  for (int b = 0; b < N_IMG; ++b) {
    const u32* ib = imgbase + (size_t)b * 8193u + cbin;
    s += ib[1] - ib[0];
  }
  total[cbin] = s;
#endif
}

__global__ __launch_bounds__(256) void k_scan1(const u32* __restrict__ total, u32* __restrict__ state) {
  __shared__ u32 psum[256];
  int t = threadIdx.x;
  u32 s = 0;
  for (int j = 0; j < 32; ++j) s += total[t * 32 + j];
  psum[t] = s;
  __syncthreads();
  if (t == 0) {
    const u32 K = K_SEL;
    u32 run = 0; int ci = 0;
    while (ci < 255 && run + psum[ci] <= K) { run += psum[ci]; ++ci; }
    u32 c = (u32)ci * 32u;
    while (c < (u32)ci * 32u + 31u && run + total[c] <= K) { run += total[c]; ++c; }
    state[0] = c;
    state[1] = K - run;
  }
}

// Rescan: 16-bit refinement histogram inside the selected level-1 bin (~8K global atomics total)
__global__ __launch_bounds__(256) void k_hist2(const float* __restrict__ in,
                                               const u32* __restrict__ state,
                                               u32* __restrict__ selL2) {
  const u32 sel1 = state[0];
  const float4* in4 = (const float4*)in;
  size_t i = (size_t)blockIdx.x * 256 + threadIdx.x;
  const size_t stride = (size_t)512 * 256;
  for (int t = 0; t < 64; ++t, i += stride) {
    float4 v = in4[i];
    u32 c, kv;
    bucket_key(v.x, c, kv); if (c == sel1) atomicAdd(&selL2[kv], 1u);
    bucket_key(v.y, c, kv); if (c == sel1) atomicAdd(&selL2[kv], 1u);
    bucket_key(v.z, c, kv); if (c == sel1) atomicAdd(&selL2[kv], 1u);
    bucket_key(v.w, c, kv); if (c == sel1) atomicAdd(&selL2[kv], 1u);
  }
}

__global__ __launch_bounds__(256) void k_scan2(const u32* __restrict__ selL2, u32* __restrict__ state) {
  __shared__ u32 psum[256];
  int t = threadIdx.x;
  u32 s = 0;
  for (int j = 0; j < 256; ++j) s += selL2[t * 256 + j];
  psum[t] = s;
  __syncthreads();
  if (t == 0) {
    const u32 K = state[1];
    u32 run = 0; int ci = 0;
    while (ci < 255 && run + psum[ci] <= K) { run += psum[ci]; ++ci; }
    u32 c = (u32)ci * 256u;
    while (c < (u32)ci * 256u + 255u && run + selL2[c] <= K) { run += selL2[c]; ++c; }
    state[2] = c;
  }
}

// Counting-sort scatter of 16-bit keys; also atomicMin to recover the exact percentile value
__global__ __launch_bounds__(256) void k_scatter(const float* __restrict__ in,
                                                 const u32* __restrict__ cb,
                                                 const u32* __restrict__ state,
                                                 u16* __restrict__ keys,
                                                 u32* __restrict__ tslot) {
  __shared__ u32 lcur[NBINS];
#if USE_ASYNC
  __shared__ __align__(16) float stage[8][2][128];
#endif
  const int blk = blockIdx.x;
  const int img = blk >> 2;
  const u32 sel1 = state[0], sel2 = state[2];
  const u32* src_cb = cb + (size_t)blk * NBINS;
  for (int i = threadIdx.x; i < NBINS; i += 256) lcur[i] = src_cb[i];
  __syncthreads();
  const u32 lane = threadIdx.x & 31u, w = threadIdx.x >> 5;
  const float* src = in + (size_t)blk * EPB + (size_t)w * 8192u;
  u16* kimg = keys + (size_t)img * IMG_E;
  auto emit = [&](float v) {
    u32 c, kv; bucket_key(v, c, kv);
    u32 pos = atomicAdd(&lcur[c], 1u);
    kimg[pos] = (u16)kv;
    if (c == sel1 && kv == sel2) atomicMin(tslot, __float_as_uint(v));
  };
#if USE_ASYNC
  {
    float* b0 = &stage[w][0][0];
    float* b1 = &stage[w][1][0];
    async_ld16(src + lane * 4u, b0 + lane * 4u);
    for (int t = 0; t < 64; ++t) {
      float* cur = (t & 1) ? b1 : b0;
      float* nxt = (t & 1) ? b0 : b1;
      if (t < 63) { async_ld16(src + (size_t)(t + 1) * 128u + lane * 4u, nxt + lane * 4u); wait_async_le<1>(); }
      else        { wait_async_le<0>(); }
      const float4 v = *(const float4*)(cur + lane * 4u);
      emit(v.x); emit(v.y); emit(v.z); emit(v.w);
    }
  }
#else
  for (int t = 0; t < 64; ++t) {
    const float4 v = *(const float4*)(src + (size_t)t * 128u + lane * 4u);
    emit(v.x); emit(v.y); emit(v.z); emit(v.w);
  }
#endif
}

// Query: out[b,i,j] = rank(in[b,2i,2j]) / 262144, clipped to 0 below threshold T
__global__ __launch_bounds__(256) void k_query(const float* __restrict__ in,
                                               const u32* __restrict__ imgbase,
                                               const u16* __restrict__ keys,
                                               const u32* __restrict__ state,
                                               float* __restrict__ out) {
  u32 t = blockIdx.x * 256u + threadIdx.x;
  u32 b = t >> 16, r = (t >> 8) & 255u, col = t & 255u;
  float v = in[((size_t)b * 512u + 2u * r) * 512u + 2u * col];
  float T = fmaxf(__uint_as_float(state[3]), 1e-3f);
  float res = 0.0f;
  if (v > T) {
    u32 c, kv; bucket_key(v, c, kv);
    const u32* ib = imgbase + (size_t)b * 8193u;
    u32 base = ib[c], end = ib[c + 1];
    const u16* kp = keys + (size_t)b * IMG_E;
    u32 cnt = base;                       // everything in lower buckets is strictly < v
    u16 kvs = (u16)kv;
    for (u32 i = base; i < end; ++i) cnt += (kp[i] < kvs) ? 1u : 0u;
    res = (float)cnt * (1.0f / 262144.0f);
  }
  out[t] = res;
}

// ---------------- host launcher ----------------
extern "C" void kernel_launch(void* const* d_in, const int* in_sizes, int n_in,
                              void* d_out, int out_size, void* d_ws, size_t ws_size,
                              hipStream_t stream) {
  const float* in = (const float*)d_in[0];
  float* out = (float*)d_out;
  char* ws = (char*)d_ws;

  u32* bh      = (u32*)(ws + WS_BH);
  u32* imgbase = (u32*)(ws + WS_IMGBASE);
  u16* keys    = (u16*)(ws + WS_KEYS);
  u32* selL2   = (u32*)(ws + WS_SELL2);
  u32* total   = (u32*)(ws + WS_TOTAL);
  u32* state   = (u32*)(ws + WS_STATE);

  k_zero   <<<256, 256, 0, stream>>>(selL2, state);
  k_hist   <<<NBLK, 256, 0, stream>>>(in, bh);
  k_prefix <<<N_IMG, 256, 0, stream>>>(bh, imgbase);
  k_sumhist<<<64, 256, 0, stream>>>(imgbase, total);
  k_scan1  <<<1, 256, 0, stream>>>(total, state);
  k_hist2  <<<512, 256, 0, stream>>>(in, state, selL2);
  k_scan2  <<<1, 256, 0, stream>>>(selL2, state);
  k_scatter<<<NBLK, 256, 0, stream>>>(in, bh, state, keys, state + 3);
  k_query  <<<OUT_E / 256, 256, 0, stream>>>(in, imgbase, keys, state, out);
}